// MultiSegmentAttention_20761871909559
// MI455X (gfx1250) — compile-verified
//
#include <hip/hip_runtime.h>

// ---------------------------------------------------------------------------
// MultiSegmentAttention for MI455X (gfx1250), fp32 end-to-end using
// V_WMMA_F32_16X16X4_F32 (full-precision matrix cores; output-bound problem:
// 302MB attn + 50MB ctx writes dominate -> ~18us HBM floor @ 23.3TB/s).
//  - S (192x192 scores) kept resident in LDS (320KB/WGP) to avoid ~900MB of
//    extra global traffic.
//  - Staging into LDS uses GLOBAL_LOAD_ASYNC_TO_LDS_B128 (ASYNCcnt path).
//  - attn/ctx outputs use non-temporal stores (write-once streams; keep the
//    192MB L2 for data that is actually re-read: x and the q workspace).
// ---------------------------------------------------------------------------

typedef __attribute__((ext_vector_type(2))) float v2f;
typedef __attribute__((ext_vector_type(8))) float v8f;

constexpr int BB = 256;    // batch
constexpr int NN = 192;    // sequence
constexpr int CC = 256;    // embed dim
constexpr int GG = 8;      // groups
constexpr int DD = 32;     // head dim
constexpr int LDT = 36;    // LDS pitch for 32-wide tiles: 144B rows (16B aligned
                           // for async B128) and conflict-free A-frag columns
                           // (gcd(36,64)=4 -> 16 distinct banks for 16 lanes)
constexpr int LDS_S = 193; // LDS pitch for 192-wide S (192%64==0 is worst case)
constexpr float SCALE_F = 0.17677669529663687f; // 1/sqrt(32)

// D = A(16x4,f32) * B(4x16,f32) + C(16x16,f32)
__device__ __forceinline__ v8f wmma4(v2f a, v2f b, v8f c) {
  // 8 args: (neg_a, A, neg_b, B, c_mod, C, reuse_a, reuse_b)
  return __builtin_amdgcn_wmma_f32_16x16x4_f32(false, a, false, b, (short)0, c,
                                               false, false);
}

// Low 32 bits of a generic pointer into LDS == wave-relative LDS byte offset
// (SHARED aperture base has zero low bits; HW truncates to LDS_ADDR.U32).
__device__ __forceinline__ unsigned lds_off_u32(const void* p) {
  return (unsigned)(unsigned long long)(uintptr_t)p;
}

// Async copy 16B memory -> LDS, tracked by ASYNCcnt (no VGPR data roundtrip).
__device__ __forceinline__ void async_copy_b128(float* lds_dst,
                                                const float* src) {
  asm volatile("global_load_async_to_lds_b128 %0, %1, off"
               :
               : "v"(lds_off_u32(lds_dst)),
                 "v"((unsigned long long)(uintptr_t)src)
               : "memory");
}

__device__ __forceinline__ void wait_asynccnt0() {
  asm volatile("s_wait_asynccnt 0" ::: "memory");
}

// A fragment: 16x4 (MxK) slice of row-major M (leading dim ld).
// lane<16: VGPR0=A[lane][k0], VGPR1=A[lane][k0+1]; lane>=16: K=k0+2,k0+3.
__device__ __forceinline__ v2f load_a(const float* M, int ld, int row0, int k0,
                                      int lane) {
  int r = row0 + (lane & 15);
  int k = k0 + ((lane >> 4) << 1);
  v2f a;
  a.x = M[(size_t)r * ld + k];
  a.y = M[(size_t)r * ld + k + 1];
  return a;
}

// B fragment: 4x16 (KxN) slice taken directly from row-major (K x N) M.
__device__ __forceinline__ v2f load_b(const float* M, int ld, int k0, int col0,
                                      int lane) {
  int c = col0 + (lane & 15);
  int k = k0 + ((lane >> 4) << 1);
  v2f b;
  b.x = M[(size_t)k * ld + c];
  b.y = M[(size_t)(k + 1) * ld + c];
  return b;
}

// B fragment: 4x16 = transpose of row-major (N x K) M, i.e. B[k][c] = M[c][k].
__device__ __forceinline__ v2f load_bT(const float* M, int ld, int k0, int col0,
                                       int lane) {
  int c = col0 + (lane & 15);
  int k = k0 + ((lane >> 4) << 1);
  v2f b;
  b.x = M[(size_t)c * ld + k];
  b.y = M[(size_t)c * ld + k + 1];
  return b;
}

// C/D 16x16: VGPR r holds row (r or r+8), col = lane&15.
__device__ __forceinline__ void store_c(float* M, int ld, int row0, int col0,
                                        int lane, v8f c) {
  int r0 = row0 + ((lane >> 4) << 3);
  int col = col0 + (lane & 15);
#pragma unroll
  for (int r = 0; r < 8; ++r) M[(size_t)(r0 + r) * ld + col] = c[r];
}

// Same but with non-temporal stores (write-once global streams).
__device__ __forceinline__ void store_c_nt(float* M, int ld, int row0, int col0,
                                           int lane, v8f c) {
  int r0 = row0 + ((lane >> 4) << 3);
  int col = col0 + (lane & 15);
#pragma unroll
  for (int r = 0; r < 8; ++r)
    __builtin_nontemporal_store(c[r], &M[(size_t)(r0 + r) * ld + col]);
}

// ---------------------------------------------------------------------------
// Kernel 1: q = x @ Wq^T + bq   ((B*N) x 256) @ (256 x 32) -> workspace
// 4 waves/block, each wave owns a 16x32 output tile (2 accumulators).
// ---------------------------------------------------------------------------
__global__ void __launch_bounds__(128) qproj_kernel(
    const float* __restrict__ x, const float* __restrict__ Wq,
    const float* __restrict__ bq, float* __restrict__ q) {
  const int lane = threadIdx.x & 31;
  const int wave = threadIdx.x >> 5;
  const int row0 = blockIdx.x * 64 + wave * 16;

  const float b0 = bq[lane & 15];
  const float b1 = bq[16 + (lane & 15)];
  v8f acc0 = {}, acc1 = {};
#pragma unroll
  for (int r = 0; r < 8; ++r) {
    acc0[r] = b0;
    acc1[r] = b1;
  }

#pragma unroll 4
  for (int k = 0; k < CC; k += 4) {
    v2f a = load_a(x, CC, row0, k, lane);
    v2f w0 = load_bT(Wq, CC, k, 0, lane);   // B = Wq^T
    acc0 = wmma4(a, w0, acc0);
    v2f w1 = load_bT(Wq, CC, k, 16, lane);
    acc1 = wmma4(a, w1, acc1);
  }
  // q workspace is re-read by the attention kernel soon: keep RT (cacheable).
  store_c(q, DD, row0, 0, lane, acc0);
  store_c(q, DD, row0, 16, lane, acc1);
}

// ---------------------------------------------------------------------------
// Kernel 2: one block per (b,g). 256 threads = 8 waves.
//   stage q,V,Wk via async-to-LDS (B128)  -> s_wait_asynccnt 0
//   k_s = v_s @ Wk[g]^T + bk[g]           (WMMA, LDS->LDS)
//   S   = q_s @ k_s^T * SCALE             (WMMA, LDS-resident 192x192)
//   P   = softmax_rows(S)                 (wave-per-row, wave32 shuffles)
//   ctx = P @ v_s                         (WMMA)
// ---------------------------------------------------------------------------
__global__ void __launch_bounds__(256) attn_kernel(
    const float* __restrict__ x, const float* __restrict__ Wk,
    const float* __restrict__ bk, const float* __restrict__ qws,
    float* __restrict__ ctx_out, float* __restrict__ attn_out) {
  extern __shared__ float smem[];
  float* q_s = smem;                 // NN * LDT
  float* k_s = q_s + NN * LDT;       // NN * LDT
  float* v_s = k_s + NN * LDT;       // NN * LDT
  float* wk_s = v_s + NN * LDT;      // DD * LDT
  float* bk_s = wk_s + DD * LDT;     // DD
  float* S = bk_s + DD;              // NN * LDS_S

  const int tid = threadIdx.x;
  const int lane = tid & 31;
  const int wave = tid >> 5;
  const int bg = blockIdx.x;
  const int b = bg >> 3;
  const int g = bg & 7;

  const float* xb = x + (size_t)b * NN * CC;
  const float* qb = qws + (size_t)b * NN * DD;

  // ---- async-stage q, v (= x group slice), Wk[g] into LDS (16B chunks) ----
  // q_s / v_s: 192 rows x 8 chunks = 1536 chunks -> 6 per thread (uniform).
  for (int c = tid; c < NN * 8; c += 256) {
    int n = c >> 3, d4 = (c & 7) << 2;
    async_copy_b128(q_s + n * LDT + d4, qb + n * DD + d4);
    async_copy_b128(v_s + n * LDT + d4, xb + (size_t)n * CC + g * DD + d4);
  }
  // Wk[g]: 32 rows x 8 chunks = 256 chunks -> 1 per thread (uniform).
  {
    int o = tid >> 3, d4 = (tid & 7) << 2;
    async_copy_b128(wk_s + o * LDT + d4, Wk + g * DD * DD + o * DD + d4);
  }
  if (tid < DD) bk_s[tid] = bk[g * DD + tid];
  wait_asynccnt0();
  __syncthreads();

  // ---- K projection: k[n][o] = sum_i v[n][i] * Wk[g][o][i] + bk[g][o] ----
  // 24 tiles (12 row-tiles x 2 col-tiles), 3 per wave.
#pragma unroll
  for (int i = 0; i < 3; ++i) {
    int t = wave + 8 * i;
    int row0 = (t >> 1) * 16, col0 = (t & 1) * 16;
    const float bias = bk_s[col0 + (lane & 15)];
    v8f acc = {};
#pragma unroll
    for (int r = 0; r < 8; ++r) acc[r] = bias;
#pragma unroll
    for (int kk = 0; kk < DD; kk += 4) {
      v2f a = load_a(v_s, LDT, row0, kk, lane);
      v2f w = load_bT(wk_s, LDT, kk, col0, lane);  // B = Wk[g]^T
      acc = wmma4(a, w, acc);
    }
    store_c(k_s, LDT, row0, col0, lane, acc);
  }
  __syncthreads();

  // ---- S = q @ k^T * SCALE : 144 tiles, 18 per wave ----
  for (int i = 0; i < 18; ++i) {
    int t = wave + 8 * i;
    int row0 = (t / 12) * 16, col0 = (t % 12) * 16;
    v8f acc = {};
#pragma unroll
    for (int kk = 0; kk < DD; kk += 4) {
      v2f a = load_a(q_s, LDT, row0, kk, lane);
      v2f kb = load_bT(k_s, LDT, kk, col0, lane);  // B = k^T
      acc = wmma4(a, kb, acc);
    }
#pragma unroll
    for (int r = 0; r < 8; ++r) acc[r] *= SCALE_F;
    store_c(S, LDS_S, row0, col0, lane, acc);
  }
  __syncthreads();

  // ---- softmax over rows of S; emit attn output (non-temporal stream) ----
  for (int r = wave; r < NN; r += 8) {
    float* Srow = S + r * LDS_S;
    float vals[6];
    float mx = -3.0e38f;
#pragma unroll
    for (int m = 0; m < 6; ++m) {
      vals[m] = Srow[lane + 32 * m];
      mx = fmaxf(mx, vals[m]);
    }
#pragma unroll
    for (int off = 16; off >= 1; off >>= 1)
      mx = fmaxf(mx, __shfl_xor(mx, off, 32));
    float sum = 0.f;
#pragma unroll
    for (int m = 0; m < 6; ++m) {
      vals[m] = __expf(vals[m] - mx);
      sum += vals[m];
    }
#pragma unroll
    for (int off = 16; off >= 1; off >>= 1) sum += __shfl_xor(sum, off, 32);
    const float inv = 1.0f / sum;
    float* arow = attn_out + ((size_t)bg * NN + r) * NN;
#pragma unroll
    for (int m = 0; m < 6; ++m) {
      float p = vals[m] * inv;
      Srow[lane + 32 * m] = p;
      __builtin_nontemporal_store(p, &arow[lane + 32 * m]);
    }
  }
  __syncthreads();

  // ---- ctx = P @ V : 24 tiles, 3 per wave; store to (b, n, g*32 + d) ----
  float* cb = ctx_out + (size_t)b * NN * CC + g * DD;
#pragma unroll
  for (int i = 0; i < 3; ++i) {
    int t = wave + 8 * i;
    int row0 = (t >> 1) * 16, col0 = (t & 1) * 16;
    v8f acc = {};
    for (int kk = 0; kk < NN; kk += 4) {
      v2f a = load_a(S, LDS_S, row0, kk, lane);
      v2f vb = load_b(v_s, LDT, kk, col0, lane);  // B = V (K x N row-major)
      acc = wmma4(a, vb, acc);
    }
    store_c_nt(cb, CC, row0, col0, lane, acc);
  }
}

// ---------------------------------------------------------------------------
extern "C" void kernel_launch(void* const* d_in, const int* in_sizes, int n_in,
                              void* d_out, int out_size, void* d_ws,
                              size_t ws_size, hipStream_t stream) {
  const float* x = (const float*)d_in[0];   // (B, N, 256)
  const float* Wq = (const float*)d_in[1];  // (32, 256)
  const float* bq = (const float*)d_in[2];  // (32,)
  const float* Wk = (const float*)d_in[3];  // (8, 32, 32)
  const float* bk = (const float*)d_in[4];  // (8, 32)

  float* ctx_out = (float*)d_out;                    // B*N*256
  float* attn_out = ctx_out + (size_t)BB * NN * CC;  // B*8*192*192
  float* qws = (float*)d_ws;                         // B*N*32 (6 MB)

  // Kernel 1: q projection into workspace (read x once here instead of 8x).
  qproj_kernel<<<(BB * NN) / 64, 128, 0, stream>>>(x, Wq, bq, qws);

  // Kernel 2: fused per-(b,g) attention; S kept in LDS (~230 KB of 320 KB/WGP).
  const size_t smem_bytes =
      (size_t)(3 * NN * LDT + DD * LDT + DD + NN * LDS_S) * sizeof(float);
  (void)hipFuncSetAttribute((const void*)attn_kernel,
                            hipFuncAttributeMaxDynamicSharedMemorySize,
                            (int)smem_bytes);
  attn_kernel<<<BB * GG, 256, smem_bytes, stream>>>(x, Wk, bk, qws, ctx_out,
                                                    attn_out);
}